// InvKin_10376640987190
// MI455X (gfx1250) — compile-verified
//
#include <hip/hip_runtime.h>
#include <math.h>

typedef __attribute__((ext_vector_type(2))) float v2f;
typedef __attribute__((ext_vector_type(8))) float v8f;

#define EPSBN   1e-5f
#define HALF_PI 1.57079632679489661923f

// workspace layout (float offsets)
#define WS_SUMX   0      // 3
#define WS_SUMXX  4      // 9 (3x3)
#define WS_W1E    16     // 64x3 folded layer-1 weights
#define WS_B1E    208    // 64
#define WS_SUMH   272    // 64
#define WS_SUMHH  336    // 64x64
#define WS_W2E    4432   // 64x64 folded
#define WS_B2E    8528   // 64
#define WS_W3E    8592   // 64x64 folded
#define WS_B3E    12688  // 64

__device__ __forceinline__ v8f wmma4(v2f a, v2f b, v8f c) {
  // V_WMMA_F32_16X16X4_F32 : D = A(16x4,f32) x B(4x16,f32) + C(16x16,f32)
  return __builtin_amdgcn_wmma_f32_16x16x4_f32(false, a, false, b, (short)0, c, false, false);
}

// ---------------- init: zero the statistic accumulators ----------------
__global__ void k_init(float* wsf) {
  int i = blockIdx.x * blockDim.x + threadIdx.x;
  if (i < WS_W2E) wsf[i] = 0.f;   // sums + (scratch) W1E/B1E region
}

// ---------------- pass 0: mean(x), E[x x^T] ----------------
__global__ void k_xstats(const float* __restrict__ x, float* wsf, int Bn) {
  float sx0 = 0.f, sx1 = 0.f, sx2 = 0.f;
  float s00 = 0.f, s01 = 0.f, s02 = 0.f, s11 = 0.f, s12 = 0.f, s22 = 0.f;
  int stride = gridDim.x * blockDim.x;
  for (int r = blockIdx.x * blockDim.x + threadIdx.x; r < Bn; r += stride) {
    float a = x[r * 3 + 0], b = x[r * 3 + 1], c = x[r * 3 + 2];
    sx0 += a; sx1 += b; sx2 += c;
    s00 += a * a; s01 += a * b; s02 += a * c;
    s11 += b * b; s12 += b * c; s22 += c * c;
  }
  __shared__ float red[9];
  if (threadIdx.x < 9) red[threadIdx.x] = 0.f;
  __syncthreads();
  atomicAdd(&red[0], sx0); atomicAdd(&red[1], sx1); atomicAdd(&red[2], sx2);
  atomicAdd(&red[3], s00); atomicAdd(&red[4], s01); atomicAdd(&red[5], s02);
  atomicAdd(&red[6], s11); atomicAdd(&red[7], s12); atomicAdd(&red[8], s22);
  __syncthreads();
  if (threadIdx.x == 0) {
    atomicAdd(&wsf[WS_SUMX + 0], red[0]);
    atomicAdd(&wsf[WS_SUMX + 1], red[1]);
    atomicAdd(&wsf[WS_SUMX + 2], red[2]);
    // full symmetric 3x3
    atomicAdd(&wsf[WS_SUMXX + 0], red[3]);
    atomicAdd(&wsf[WS_SUMXX + 1], red[4]);
    atomicAdd(&wsf[WS_SUMXX + 2], red[5]);
    atomicAdd(&wsf[WS_SUMXX + 3], red[4]);
    atomicAdd(&wsf[WS_SUMXX + 4], red[6]);
    atomicAdd(&wsf[WS_SUMXX + 5], red[7]);
    atomicAdd(&wsf[WS_SUMXX + 6], red[5]);
    atomicAdd(&wsf[WS_SUMXX + 7], red[7]);
    atomicAdd(&wsf[WS_SUMXX + 8], red[8]);
  }
}

// ---------------- fold BN1 into W1eff/b1eff ----------------
__global__ void k_fin1(const float* __restrict__ W1, const float* __restrict__ b1,
                       const float* __restrict__ g1, const float* __restrict__ be1,
                       float* wsf, float invB) {
  int j = threadIdx.x;  // 64 threads
  float mu[3];
  mu[0] = wsf[WS_SUMX + 0] * invB;
  mu[1] = wsf[WS_SUMX + 1] * invB;
  mu[2] = wsf[WS_SUMX + 2] * invB;
  float C[3][3];
#pragma unroll
  for (int a = 0; a < 3; ++a)
#pragma unroll
    for (int b = 0; b < 3; ++b)
      C[a][b] = wsf[WS_SUMXX + a * 3 + b] * invB - mu[a] * mu[b];
  float w[3] = {W1[j * 3 + 0], W1[j * 3 + 1], W1[j * 3 + 2]};
  float t = w[0] * mu[0] + w[1] * mu[1] + w[2] * mu[2];
  float mean = t + b1[j];
  float var = 0.f;
#pragma unroll
  for (int a = 0; a < 3; ++a)
#pragma unroll
    for (int b = 0; b < 3; ++b)
      var = fmaf(w[a] * C[a][b], w[b], var);
  float s = g1[j] * rsqrtf(var + EPSBN);
  wsf[WS_W1E + j * 3 + 0] = w[0] * s;
  wsf[WS_W1E + j * 3 + 1] = w[1] * s;
  wsf[WS_W1E + j * 3 + 2] = w[2] * s;
  wsf[WS_B1E + j] = (b1[j] - mean) * s + be1[j];
}

// ---------------- fold BN2/BN3 from Σh, Σhh^T ----------------
__global__ void k_finmid(const float* __restrict__ W, const float* __restrict__ b,
                         const float* __restrict__ g, const float* __restrict__ be,
                         float* wsf, int effW, int effB, float invB) {
  __shared__ float sHH[4096];
  __shared__ float sWT[4096];  // transposed weight: sWT[k*64+j] = W[j][k]
  __shared__ float mu[64];
  int tid = threadIdx.x;  // 64 threads
  for (int i = tid; i < 4096; i += 64) {
    sHH[i] = wsf[WS_SUMHH + i];
    int jj = i & 63, ll = i >> 6;
    sWT[i] = W[jj * 64 + ll];  // sWT[ll*64+jj]
  }
  mu[tid] = wsf[WS_SUMH + tid] * invB;
  __syncthreads();
  // reset global stat accumulators for the next pass
  for (int i = tid; i < 4096; i += 64) wsf[WS_SUMHH + i] = 0.f;
  wsf[WS_SUMH + tid] = 0.f;

  int j = tid;
  float t = 0.f;
  for (int k = 0; k < 64; ++k) t = fmaf(sWT[k * 64 + j], mu[k], t);
  float mean = t + b[j];
  float q = 0.f;
  for (int k = 0; k < 64; ++k) {
    float inner = 0.f;
    for (int l = 0; l < 64; ++l) inner = fmaf(sHH[k * 64 + l], sWT[l * 64 + j], inner);
    q = fmaf(sWT[k * 64 + j], inner, q);
  }
  float var = q * invB - t * t;  // W_j Cov W_j^T
  float s = g[j] * rsqrtf(var + EPSBN);
  for (int k = 0; k < 64; ++k) wsf[effW + j * 64 + k] = sWT[k * 64 + j] * s;
  wsf[effB + j] = (b[j] - mean) * s + be[j];
}

// ---------------- forward kinematics helpers ----------------
__device__ __forceinline__ void dh4(float d, float th, float r, float al, float* m) {
  float ct = cosf(th), st = sinf(th);
  float ca = cosf(al), sa = sinf(al);
  m[0] = ct; m[1] = -st * ca; m[2] = st * sa;  m[3] = r * ct;
  m[4] = st; m[5] = ct * ca;  m[6] = -ct * sa; m[7] = r * st;
  m[8] = 0.f; m[9] = sa; m[10] = ca; m[11] = d;
  m[12] = 0.f; m[13] = 0.f; m[14] = 0.f; m[15] = 1.f;
}
__device__ __forceinline__ void mm4(const float* a, const float* b, float* c) {
#pragma unroll
  for (int i = 0; i < 4; ++i)
#pragma unroll
    for (int j = 0; j < 4; ++j) {
      float s = 0.f;
#pragma unroll
      for (int k = 0; k < 4; ++k) s = fmaf(a[i * 4 + k], b[k * 4 + j], s);
      c[i * 4 + j] = s;
    }
}

// WMMA GEMM: h(16x64,in LDS, wave-private) <- relu(h @ Wl^T + bl), Wl,bl in LDS
__device__ __forceinline__ void gemm_relu(float* h, const float* Wl, const float* bl,
                                          int half, int n16) {
  const v8f zero8 = {0.f, 0.f, 0.f, 0.f, 0.f, 0.f, 0.f, 0.f};
  v8f c0 = zero8, c1 = zero8, c2 = zero8, c3 = zero8;
#pragma unroll
  for (int kk = 0; kk < 64; kk += 4) {
    int k0 = kk + 2 * half;
    v2f a = *reinterpret_cast<const v2f*>(&h[n16 * 64 + k0]);
    v2f b0 = *reinterpret_cast<const v2f*>(&Wl[(0 * 16 + n16) * 64 + k0]);
    v2f b1 = *reinterpret_cast<const v2f*>(&Wl[(1 * 16 + n16) * 64 + k0]);
    v2f b2 = *reinterpret_cast<const v2f*>(&Wl[(2 * 16 + n16) * 64 + k0]);
    v2f b3 = *reinterpret_cast<const v2f*>(&Wl[(3 * 16 + n16) * 64 + k0]);
    c0 = wmma4(a, b0, c0);
    c1 = wmma4(a, b1, c1);
    c2 = wmma4(a, b2, c2);
    c3 = wmma4(a, b3, c3);
  }
#pragma unroll
  for (int v = 0; v < 8; ++v) {
    int m = v + 8 * half;
    h[m * 64 + 0 * 16 + n16] = fmaxf(c0[v] + bl[0 * 16 + n16], 0.f);
    h[m * 64 + 1 * 16 + n16] = fmaxf(c1[v] + bl[1 * 16 + n16], 0.f);
    h[m * 64 + 2 * 16 + n16] = fmaxf(c2[v] + bl[2 * 16 + n16], 0.f);
    h[m * 64 + 3 * 16 + n16] = fmaxf(c3[v] + bl[3 * 16 + n16], 0.f);
  }
}

// ---------------- main pass kernel ----------------
// MODE 1: compute h1, accumulate Σh1, Σh1h1^T
// MODE 2: compute h1->h2, accumulate Σh2, Σh2h2^T
// MODE 3: compute h1->h2->h3->thetas->FK, write outputs
//
// __launch_bounds__(128, 1): LDS (~70KB/block) already caps us at ~4 blocks/WGP,
// so let the compiler use a large VGPR allocation instead of spilling the
// Gram-matrix accumulators to scratch.
template <int MODE>
__global__ __launch_bounds__(128, 1) void k_pass(
    const float* __restrict__ x, float* wsf,
    const float* __restrict__ W4, const float* __restrict__ b4,
    float* __restrict__ out, int Bn, int numTiles) {
  extern __shared__ float smem[];
  float* sW1e  = smem;            // 192
  float* sb1e  = smem + 192;      // 64
  float* sW2   = smem + 256;      // 4096
  float* sb2   = smem + 4352;     // 64
  float* sW3   = smem + 4416;     // 4096
  float* sb3   = smem + 8512;     // 64
  float* sW4   = smem + 8576;     // 192
  float* sX    = smem + 8768;     // 192
  float* accS  = smem + 8960;     // 64
  float* accHH = smem + 9024;     // 4096
  float* hb    = smem + 13120;    // 4 waves * 1024
  float* thT   = smem + 17216;    // 4 waves * 48

  const int tid = threadIdx.x;
  for (int i = tid; i < 192; i += 128) sW1e[i] = wsf[WS_W1E + i];
  for (int i = tid; i < 64; i += 128) sb1e[i] = wsf[WS_B1E + i];
  if (MODE >= 2) {
    for (int i = tid; i < 4096; i += 128) sW2[i] = wsf[WS_W2E + i];
    for (int i = tid; i < 64; i += 128) sb2[i] = wsf[WS_B2E + i];
  }
  if (MODE == 3) {
    for (int i = tid; i < 4096; i += 128) sW3[i] = wsf[WS_W3E + i];
    for (int i = tid; i < 64; i += 128) sb3[i] = wsf[WS_B3E + i];
    for (int i = tid; i < 192; i += 128) sW4[i] = W4[i];
  } else {
    for (int i = tid; i < 64; i += 128) accS[i] = 0.f;
    for (int i = tid; i < 4096; i += 128) accHH[i] = 0.f;
  }
  __syncthreads();

  const int w = tid >> 5;
  const int lane = tid & 31;
  const int half = lane >> 4;
  const int n16 = lane & 15;
  float* h = hb + w * 1024;

  const v8f zero8 = {0.f, 0.f, 0.f, 0.f, 0.f, 0.f, 0.f, 0.f};
  // Upper-triangle Gram tiles only (Σ h^T h is symmetric): (mt,nt), mt<=nt.
  // 10 tiles * 8 VGPR = 80 VGPRs instead of 128.
  v8f acc[10];
  float s0 = 0.f, s1 = 0.f;
  if (MODE <= 2) {
#pragma unroll
    for (int i = 0; i < 10; ++i) acc[i] = zero8;
  }
  float b40 = 0.f, b41 = 0.f, b42 = 0.f;
  if (MODE == 3) { b40 = b4[0]; b41 = b4[1]; b42 = b4[2]; }

  for (int tile = blockIdx.x; tile < numTiles; tile += gridDim.x) {
    const int rowBase = tile * 64;
    for (int i = tid; i < 192; i += 128) {
      int r = rowBase + i / 3;
      sX[i] = (r < Bn) ? x[r * 3 + (i % 3)] : 0.f;
    }
    __syncthreads();
    const int wRowBase = rowBase + w * 16;

    // ---- layer 1 (BN folded): h1 = relu(x @ W1eff^T + b1eff), wave-private tile
#pragma unroll 4
    for (int r = 0; r < 16; ++r) {
      float x0 = sX[(w * 16 + r) * 3 + 0];
      float x1 = sX[(w * 16 + r) * 3 + 1];
      float x2 = sX[(w * 16 + r) * 3 + 2];
      int f = lane;
      float v = fmaf(x0, sW1e[f * 3], fmaf(x1, sW1e[f * 3 + 1], fmaf(x2, sW1e[f * 3 + 2], sb1e[f])));
      h[r * 64 + f] = fmaxf(v, 0.f);
      f = lane + 32;
      v = fmaf(x0, sW1e[f * 3], fmaf(x1, sW1e[f * 3 + 1], fmaf(x2, sW1e[f * 3 + 2], sb1e[f])));
      h[r * 64 + f] = fmaxf(v, 0.f);
    }
    if (wRowBase + 16 > Bn) {  // zero invalid tail rows so stats stay exact
      for (int r = 0; r < 16; ++r)
        if (wRowBase + r >= Bn) { h[r * 64 + lane] = 0.f; h[r * 64 + lane + 32] = 0.f; }
    }

    if (MODE >= 2) {
      gemm_relu(h, sW2, sb2, half, n16);
      if (wRowBase + 16 > Bn) {
        for (int r = 0; r < 16; ++r)
          if (wRowBase + r >= Bn) { h[r * 64 + lane] = 0.f; h[r * 64 + lane + 32] = 0.f; }
      }
    }

    if (MODE == 3) {
      gemm_relu(h, sW3, sb3, half, n16);
      // thetas = h3 @ W4^T + b4 via WMMA with zero-padded 16-feature W4
      v8f ct = zero8;
#pragma unroll
      for (int kk = 0; kk < 64; kk += 4) {
        int k0 = kk + 2 * half;
        v2f a = *reinterpret_cast<const v2f*>(&h[n16 * 64 + k0]);
        v2f b;
        b.x = (n16 < 3) ? sW4[n16 * 64 + k0] : 0.f;
        b.y = (n16 < 3) ? sW4[n16 * 64 + k0 + 1] : 0.f;
        ct = wmma4(a, b, ct);
      }
#pragma unroll
      for (int v = 0; v < 8; ++v) {
        int rl = v + 8 * half;
        if (n16 < 3) {
          float bb = (n16 == 0) ? b40 : ((n16 == 1) ? b41 : b42);
          float t = ct[v] + bb;
          thT[w * 48 + rl * 3 + n16] = t;
          int gr = wRowBase + rl;
          if (gr < Bn) out[gr * 3 + n16] = t;  // output 0: thetas
        }
      }
      // FK epilogue: one row per lane (0..15); same-wave LDS ops are in-order
      if (lane < 16) {
        int gr = wRowBase + lane;
        if (gr < Bn) {
          float t0 = thT[w * 48 + lane * 3 + 0];
          float t1 = thT[w * 48 + lane * 3 + 1];
          float t2 = thT[w * 48 + lane * 3 + 2];
          float M0[16], M1[16], M2[16], P[16], Q[16];
          dh4(0.f, t0, 0.f, HALF_PI, M0);
          dh4(0.f, t1, 0.12f, 0.f, M1);
          dh4(0.f, t2, 0.115f, 0.f, M2);
          mm4(M0, M1, P);
          mm4(P, M2, Q);
          size_t base = (size_t)Bn * 3 + (size_t)gr * 3;
          out[base + 0] = Q[3];
          out[base + 1] = Q[7];
          out[base + 2] = Q[11];
        }
      }
    } else {
      // ---- stats: Σh via scalar adds, Σ h^T h via WMMA rank-16 update.
      // Only upper-triangle tiles (mt<=nt); mirrored at flush.
#pragma unroll
      for (int kk = 0; kk < 16; kk += 4) {
        int k0 = kk + 2 * half;
        v2f op[4];
#pragma unroll
        for (int t = 0; t < 4; ++t) {
          op[t].x = h[(k0 + 0) * 64 + t * 16 + n16];
          op[t].y = h[(k0 + 1) * 64 + t * 16 + n16];
        }
        int idx = 0;
#pragma unroll
        for (int mt = 0; mt < 4; ++mt)
#pragma unroll
          for (int nt = mt; nt < 4; ++nt) {
            acc[idx] = wmma4(op[mt], op[nt], acc[idx]);
            ++idx;
          }
      }
      for (int r = 0; r < 16; ++r) { s0 += h[r * 64 + lane]; s1 += h[r * 64 + lane + 32]; }
    }
    __syncthreads();
  }

  if (MODE <= 2) {  // flush per-wave accumulators: LDS combine, then global atomics
    atomicAdd(&accS[lane], s0);
    atomicAdd(&accS[lane + 32], s1);
    int idx = 0;
#pragma unroll
    for (int mt = 0; mt < 4; ++mt)
#pragma unroll
      for (int nt = mt; nt < 4; ++nt) {
#pragma unroll
        for (int v = 0; v < 8; ++v) {
          int M = mt * 16 + v + 8 * half;
          int N = nt * 16 + n16;
          float val = acc[idx][v];
          atomicAdd(&accHH[M * 64 + N], val);
          if (mt != nt) atomicAdd(&accHH[N * 64 + M], val);  // symmetric mirror
        }
        ++idx;
      }
    __syncthreads();
    for (int i = tid; i < 64; i += 128) atomicAdd(&wsf[WS_SUMH + i], accS[i]);
    for (int i = tid; i < 4096; i += 128) atomicAdd(&wsf[WS_SUMHH + i], accHH[i]);
  }
}

extern "C" void kernel_launch(void* const* d_in, const int* in_sizes, int n_in,
                              void* d_out, int out_size, void* d_ws, size_t ws_size,
                              hipStream_t stream) {
  const float* x   = (const float*)d_in[0];
  const float* W1  = (const float*)d_in[1];
  const float* b1  = (const float*)d_in[2];
  const float* g1  = (const float*)d_in[3];
  const float* be1 = (const float*)d_in[4];
  const float* W2  = (const float*)d_in[5];
  const float* b2  = (const float*)d_in[6];
  const float* g2  = (const float*)d_in[7];
  const float* be2 = (const float*)d_in[8];
  const float* W3  = (const float*)d_in[9];
  const float* b3  = (const float*)d_in[10];
  const float* g3  = (const float*)d_in[11];
  const float* be3 = (const float*)d_in[12];
  const float* W4  = (const float*)d_in[13];
  const float* b4  = (const float*)d_in[14];
  float* out = (float*)d_out;
  float* wsf = (float*)d_ws;

  int Bn = in_sizes[0] / 3;
  float invB = 1.0f / (float)Bn;
  int numTiles = (Bn + 63) / 64;
  int passBlocks = numTiles < 2048 ? numTiles : 2048;
  size_t smemBytes = 17408 * sizeof(float);  // 69,632 B < 320 KB/WGP

  hipLaunchKernelGGL(k_init, dim3((WS_W2E + 255) / 256), dim3(256), 0, stream, wsf);
  hipLaunchKernelGGL(k_xstats, dim3(1024), dim3(256), 0, stream, x, wsf, Bn);
  hipLaunchKernelGGL(k_fin1, dim3(1), dim3(64), 0, stream, W1, b1, g1, be1, wsf, invB);
  hipLaunchKernelGGL((k_pass<1>), dim3(passBlocks), dim3(128), smemBytes, stream,
                     x, wsf, W4, b4, out, Bn, numTiles);
  hipLaunchKernelGGL(k_finmid, dim3(1), dim3(64), 0, stream, W2, b2, g2, be2, wsf,
                     WS_W2E, WS_B2E, invB);
  hipLaunchKernelGGL((k_pass<2>), dim3(passBlocks), dim3(128), smemBytes, stream,
                     x, wsf, W4, b4, out, Bn, numTiles);
  hipLaunchKernelGGL(k_finmid, dim3(1), dim3(64), 0, stream, W3, b3, g3, be3, wsf,
                     WS_W3E, WS_B3E, invB);
  hipLaunchKernelGGL((k_pass<3>), dim3(passBlocks), dim3(128), smemBytes, stream,
                     x, wsf, W4, b4, out, Bn, numTiles);
}